// Sampler_61203283968047
// MI455X (gfx1250) — compile-verified
//
#include <hip/hip_runtime.h>
#include <stdint.h>

// ---------------------------------------------------------------------------
// Sampler: temperature scale -> suppress -> top-k mask -> softmax -> inverse
// CDF sample.  One block per row; memory-bound (155 MB logits; tensor fits in
// the 192 MB L2, so only pass 0 touches HBM: ~6.7 us floor at 23.3 TB/s).
//
// CDNA5 specifics:
//  - wave32 shuffles/ballot for reductions, scans and the CDF search
//  - lane-replicated, bank-swizzled LDS histogram: conflict-free ds atomics
//  - global_load_async_to_lds_b128 + s_wait_asynccnt double buffering
//  - global_prefetch_b8
// Radix select runs on RAW logit bits (x/T, T>0, is monotone), so the hot
// loops carry no multiplies / suppress compares.
// ---------------------------------------------------------------------------

#define TPB 256
#define NWAVES (TPB / 32)
#define CHUNK 4096            // floats per LDS-staged chunk (16 KB)
#define MAXCHUNKS 64          // supports V up to 262144
#define MAXSUP 8
#define KEY_NEG_INF 0x007FFFFFu   // mono_key(-inf)

__device__ __forceinline__ uint32_t mono_key(float x) {
    const uint32_t u = __float_as_uint(x);
    return u ^ ((uint32_t)(((int32_t)u) >> 31) | 0x80000000u);
}
__device__ __forceinline__ float inv_key(uint32_t k) {
    const uint32_t u = (k & 0x80000000u) ? (k & 0x7FFFFFFFu) : ~k;
    return __uint_as_float(u);
}

// swizzled index into the lane-replicated histogram: copy c, bucket b
__device__ __forceinline__ int hidx(int c, uint32_t b) {
    return (c << 8) + (int)((b + 2u * (uint32_t)c) & 255u);
}

__global__ __launch_bounds__(TPB)
void Sampler_61203283968047_kernel(const float* __restrict__ logits,
                                   const float* __restrict__ temps,
                                   const float* __restrict__ uvec,
                                   const int* __restrict__ suppress,
                                   const int* __restrict__ topk_p,
                                   int* __restrict__ out,
                                   int V, int n_sup) {
    const int row  = blockIdx.x;
    const int tid  = threadIdx.x;
    const int lane = tid & 31;
    const int wave = tid >> 5;

    const float invT = 1.0f / temps[row];
    const int   K    = topk_p[0];
    const float* rowp = logits + (size_t)row * (size_t)V;

    // 32 KB arena: lane-replicated histogram (phase 1) / async double buffer
    // (phase 2).  uint4 declaration guarantees 16 B alignment.
    __shared__ uint4    s_arena[2048];
    uint32_t* hist2 = (uint32_t*)s_arena;            // 32 copies x 256 buckets
    float*    lbuf  = (float*)s_arena;               // 2 x CHUNK floats

    __shared__ uint32_t red_u[NWAVES];
    __shared__ uint32_t maxu[NWAVES];
    __shared__ float    redf[NWAVES];
    __shared__ float    chunkSums[MAXCHUNKS];
    __shared__ uint32_t s_prefix;
    __shared__ uint32_t s_krem;
    __shared__ float    s_rawmax;
    __shared__ float    s_base, s_target;
    __shared__ int      s_chunk;

    // suppress ids in registers (cols >= 0 so -1 never matches)
    int supreg[MAXSUP];
    const int ns = n_sup < MAXSUP ? n_sup : MAXSUP;
    int supMin = 0x7FFFFFFF, supMax = -0x7FFFFFFF;
    for (int j = 0; j < MAXSUP; ++j) {
        supreg[j] = (j < ns) ? suppress[j] : -1;
        if (j < ns) { supMin = min(supMin, supreg[j]); supMax = max(supMax, supreg[j]); }
    }

    if (tid == 0) { s_prefix = 0u; s_krem = (uint32_t)K; }

    const int niter = (V + TPB * 4 - 1) / (TPB * 4);   // uniform trip count

    // ============== Phase 1: radix-select the K-th largest raw value =======
    uint32_t kmax = 0u;
    for (int pass = 0; pass < 4; ++pass) {
        const int shift = 24 - pass * 8;
        {   // zero the replicated histogram (8 x b128 stores / thread)
            uint4* z = (uint4*)s_arena;
            for (int w = tid; w < 2048; w += TPB) z[w] = make_uint4(0u, 0u, 0u, 0u);
        }
        __syncthreads();
        const uint32_t kr    = s_krem;
        const uint32_t pref  = s_prefix;
        const uint32_t pmask = (pass == 0) ? 0u : (0xFFFFFFFFu << (shift + 8));
        const uint32_t prefm = pref & pmask;

        for (int it = 0; it < niter; ++it) {
            const int i = it * TPB * 4 + tid * 4;
            __builtin_prefetch(rowp + i + TPB * 4 * 8, 0, 1);
            float vv[4];
            if (i + 3 < V) {
                const float4 v4 = *(const float4*)(rowp + i);
                vv[0] = v4.x; vv[1] = v4.y; vv[2] = v4.z; vv[3] = v4.w;
            } else {
#pragma unroll
                for (int e = 0; e < 4; ++e) vv[e] = (i + e < V) ? rowp[i + e] : 0.f;
            }
            uint32_t key[4];
#pragma unroll
            for (int e = 0; e < 4; ++e) key[e] = mono_key(vv[e]);
            if (i <= supMax && i + 3 >= supMin) {      // rare: suppress fix
#pragma unroll
                for (int e = 0; e < 4; ++e)
#pragma unroll
                    for (int j = 0; j < MAXSUP; ++j)
                        if (i + e == supreg[j]) key[e] = KEY_NEG_INF;
            }
#pragma unroll
            for (int e = 0; e < 4; ++e) {
                const bool inb = (i + e < V);
                if (pass == 0 && inb) kmax = max(kmax, key[e]);
                if (inb && ((key[e] & pmask) == prefm))
                    atomicAdd(&hist2[hidx(lane, (key[e] >> shift) & 255u)], 1u);
            }
        }
        __syncthreads();                               // atomics complete

        // merge: thread t owns bucket t, sums its 32 lane-copies
        uint32_t cnt = 0u;
#pragma unroll
        for (int c = 0; c < 32; ++c) cnt += hist2[hidx(c, (uint32_t)tid)];

        if (pass == 0) {                               // row max (as key)
#pragma unroll
            for (int off = 16; off; off >>= 1)
                kmax = max(kmax, (uint32_t)__shfl_xor((int)kmax, off, 32));
            if (lane == 0) maxu[wave] = kmax;
        }

        // suffix sum S(tid) = sum of cnt over buckets >= tid
        uint32_t x = cnt;
#pragma unroll
        for (int off = 1; off < 32; off <<= 1) {
            const uint32_t t = (uint32_t)__shfl_down((int)x, off, 32);
            if (lane + off < 32) x += t;
        }
        if (lane == 0) red_u[wave] = x;                // wave bucket-range total
        __syncthreads();
        uint32_t hi = 0u;
        for (int w = wave + 1; w < NWAVES; ++w) hi += red_u[w];
        const uint32_t S     = x + hi;
        const uint32_t above = S - cnt;
        if (above < kr && S >= kr) {                   // unique winning bucket
            s_prefix = pref | ((uint32_t)tid << shift);
            s_krem   = kr - above;
        }
        if (pass == 0 && tid == 0) {
            uint32_t m = maxu[0];
            for (int w = 1; w < NWAVES; ++w) m = max(m, maxu[w]);
            s_rawmax = inv_key(m);
        }
        __syncthreads();
    }
    const float vkraw  = inv_key(s_prefix);            // raw-domain threshold
    const float rawmax = s_rawmax;
    const float rmaxs  = rawmax * invT;                // scaled max

    // ============== Phase 2: exp-sums via async-LDS double buffering =======
    const int nchunks = (V + CHUNK - 1) / CHUNK;

    auto stage = [&](int c) {
        const int base = c * CHUNK;
        float* dst = lbuf + (c & 1) * CHUNK;
#pragma unroll
        for (int r = 0; r < CHUNK / (TPB * 4); ++r) {  // 4 async ops / wave
            const int j = r * TPB * 4 + tid * 4;
            int col = base + j;
            if (col > V - 4) col = V - 4;              // clamp: uniform op count
            const uint64_t ga = (uint64_t)(uintptr_t)(rowp + col);
            const uint32_t la = (uint32_t)(uintptr_t)(dst + j);
            asm volatile("global_load_async_to_lds_b128 %0, %1, off"
                         :: "v"(la), "v"(ga) : "memory");
        }
    };

    stage(0);
    for (int c = 0; c < nchunks; ++c) {
        if (c + 1 < nchunks) {
            stage(c + 1);
            asm volatile("s_wait_asynccnt 4" ::: "memory");   // chunk c done
        } else {
            asm volatile("s_wait_asynccnt 0" ::: "memory");
        }
        __syncthreads();
        const int base = c * CHUNK;
        const int len  = (V - base < CHUNK) ? (V - base) : CHUNK;
        const float* lb = lbuf + (c & 1) * CHUNK;
        float ls = 0.f;
        for (int j = tid * 4; j < len; j += TPB * 4) {
            float vv[4];
            if (j + 3 < len) {                         // ds_load_b128
                const float4 v4 = *(const float4*)(lb + j);
                vv[0] = v4.x; vv[1] = v4.y; vv[2] = v4.z; vv[3] = v4.w;
            } else {
#pragma unroll
                for (int e = 0; e < 4; ++e) vv[e] = (j + e < len) ? lb[j + e] : 0.f;
            }
            const int col0 = base + j;
            if (col0 <= supMax && col0 + 3 >= supMin) {
#pragma unroll
                for (int e = 0; e < 4; ++e)
#pragma unroll
                    for (int jj = 0; jj < MAXSUP; ++jj)
                        if (col0 + e == supreg[jj]) vv[e] = -__builtin_inff();
            }
#pragma unroll
            for (int e = 0; e < 4; ++e)
                if (j + e < len && vv[e] >= vkraw)
                    ls += expf(vv[e] * invT - rmaxs);
        }
#pragma unroll
        for (int off = 16; off; off >>= 1) ls += __shfl_xor(ls, off, 32);
        if (lane == 0) redf[wave] = ls;
        __syncthreads();
        if (tid == 0) {
            float s = 0.f;
            for (int w = 0; w < NWAVES; ++w) s += redf[w];
            chunkSums[c] = s;
        }
        __syncthreads();
    }

    // ============== Phase 3: locate crossing chunk =========================
    if (tid == 0) {
        float tot = 0.f;
        for (int c = 0; c < nchunks; ++c) tot += chunkSums[c];
        const float tgt = uvec[row] * tot;
        float cum = 0.f;
        int   sel = nchunks - 1;
        float bas = tot - chunkSums[nchunks - 1];
        for (int c = 0; c < nchunks; ++c) {
            if (cum + chunkSums[c] >= tgt) { sel = c; bas = cum; break; }
            cum += chunkSums[c];
        }
        s_chunk = sel; s_base = bas; s_target = tgt;
    }
    __syncthreads();

    // ============== Phase 4: wave0 scans the chosen chunk ==================
    if (wave == 0) {
        const int   c     = s_chunk;
        float       basev = s_base;
        const float tgt   = s_target;
        const int   start = c * CHUNK;
        const int   end   = (start + CHUNK < V) ? (start + CHUNK) : V;
        int found = -1;
        for (int g = start; g < end; g += 32) {
            const int i = g + lane;
            float e = 0.f;
            if (i < end) {
                float x = rowp[i];
                if (i >= supMin && i <= supMax) {
#pragma unroll
                    for (int j = 0; j < MAXSUP; ++j)
                        if (i == supreg[j]) x = -__builtin_inff();
                }
                if (x >= vkraw) e = expf(x * invT - rmaxs);
            }
            float p = e;                               // inclusive wave32 scan
#pragma unroll
            for (int off = 1; off < 32; off <<= 1) {
                const float t = __shfl_up(p, off, 32);
                if (lane >= off) p += t;
            }
            const unsigned long long m = __ballot(basev + p >= tgt);
            if (m) { found = g + (__ffsll(m) - 1); break; }
            basev += __shfl(p, 31, 32);
        }
        if (lane == 0) out[row] = (found >= 0) ? found : (V - 1);
    }
}

extern "C" void kernel_launch(void* const* d_in, const int* in_sizes, int n_in,
                              void* d_out, int out_size, void* d_ws, size_t ws_size,
                              hipStream_t stream) {
    const float* logits   = (const float*)d_in[0];
    const float* temps    = (const float*)d_in[1];
    const float* uvec     = (const float*)d_in[2];
    const int*   suppress = (const int*)d_in[3];
    const int*   topk     = (const int*)d_in[4];

    const int B = in_sizes[1];               // temperatures: [B]
    const int V = in_sizes[0] / B;           // logits: [B, V]
    const int n_sup = in_sizes[3];

    int* out = (int*)d_out;                  // reference output dtype: int32 tokens

    Sampler_61203283968047_kernel<<<B, TPB, 0, stream>>>(
        logits, temps, uvec, suppress, topk, out, V, n_sup);
}